// DRCClsRendering_33371895890486
// MI455X (gfx1250) — compile-verified
//
#include <hip/hip_runtime.h>
#include <hip/hip_bf16.h>

// ---------------------------------------------------------------------------
// DRC classification rendering for MI455X (gfx1250, wave32).
//
// Roofline: ~0.6 GFLOP vs 33.5M scattered 64B gathers; volume (padded, 37MB)
// is L2-resident (192MB L2) -> L2-gather/latency bound. Strategy:
//   1) pack_kernel: sigmoid + zero-pad + channel-pad 13->16 so each voxel is
//      one aligned 64B cacheline (4x float4 loads with immediate offsets).
//   2) rdir_wmma_kernel: the reference's einsum 'bij,shwj->bshwi' factorizes
//      to rdir = R*(u,v,1) per ray -> computed with V_WMMA_F32_16X16X4_F32.
//   3) raymarch_kernel: 8 segments x 16 samples per ray (8192 waves for
//      latency hiding); exact cross-segment transmittance combine in LDS;
//      global_prefetch of next sample's cell.
// ---------------------------------------------------------------------------

#define BATCH   2
#define VD      64            // voxel dim
#define PD      66            // padded dim
#define NCH     13            // channels
#define PCH     16            // padded channels (64B per voxel)
#define HW      128           // image H == W
#define CNTS    128           // samples per ray
#define SEGS    8
#define SEG_LEN 16
#define NPIX    (BATCH * HW * HW)                // 32768
#define INV_FOCAL 0.0066061440427837784f         // tan(0.4)/64
#define T_NEAR  0.9f
#define T_STEP  (1.3f / 127.0f)

typedef float v2f __attribute__((ext_vector_type(2)));
typedef float v8f __attribute__((ext_vector_type(8)));

// ---------------------------------------------------------------------------
// Stage 1: vol = sigmoid(vox), zero-pad spatially by 1, pad channels to 16.
// Output layout: packed[b][z][y][x][16], 64B per voxel, cacheline aligned.
// ---------------------------------------------------------------------------
__global__ void pack_kernel(const float* __restrict__ vox,
                            float* __restrict__ packed, int total) {
    int tid = blockIdx.x * blockDim.x + threadIdx.x;
    if (tid >= total) return;
    int x  = tid % PD;
    int t1 = tid / PD;
    int y  = t1 % PD;
    int t2 = t1 / PD;
    int z  = t2 % PD;
    int b  = t2 / PD;

    float vals[PCH];
#pragma unroll
    for (int i = 0; i < PCH; ++i) vals[i] = 0.0f;

    if (x >= 1 && x <= VD && y >= 1 && y <= VD && z >= 1 && z <= VD) {
        const float* src = vox +
            ((((size_t)b * VD + (z - 1)) * VD + (y - 1)) * VD + (x - 1)) * NCH;
#pragma unroll
        for (int c = 0; c < NCH; ++c)
            vals[c] = 1.0f / (1.0f + expf(-src[c]));
    }
    float4* dst = (float4*)packed + (size_t)tid * 4;
    dst[0] = make_float4(vals[0],  vals[1],  vals[2],  vals[3]);
    dst[1] = make_float4(vals[4],  vals[5],  vals[6],  vals[7]);
    dst[2] = make_float4(vals[8],  vals[9],  vals[10], vals[11]);
    dst[3] = make_float4(vals[12], vals[13], vals[14], vals[15]);
}

// ---------------------------------------------------------------------------
// Stage 2: rdir[b,h,w,:] = R_b * (u,v,1) via V_WMMA_F32_16X16X4_F32.
// One wave handles 16 rays: A = 16x4 (rows = rays, K = dir components),
// B = 4x16 with B[k][n] = R[n][k] for n<3, else 0.  D[m][n] = rdir_n(ray m).
// A layout (ISA 7.12.2): lanes 0-15 hold K=0 (v0), K=1 (v1); lanes 16-31
// hold K=2 (v0), K=3 (v1).  B/D: row striped across lanes within a VGPR,
// halves split across lane groups (D: VGPR r = rows r and r+8).
// ---------------------------------------------------------------------------
__global__ void rdir_wmma_kernel(const float* __restrict__ cam,
                                 float* __restrict__ rdirs) {
    const int lane = threadIdx.x & 31;
    const int wib  = threadIdx.x >> 5;
    const int wave = blockIdx.x * (blockDim.x >> 5) + wib;   // 0..2047
    const int b    = wave >> 10;                             // 1024 waves / image
    const int ray0 = (wave & 1023) << 4;                     // 16 rays per wave
    const int half = lane >> 4;
    const int nl   = lane & 15;

    // A: this lane's ray is m = nl
    int ray = ray0 + nl;
    int h = ray >> 7, w = ray & 127;
    float u = ((float)w + 0.5f - 64.0f) * INV_FOCAL;
    float v = ((float)h + 0.5f - 64.0f) * INV_FOCAL;
    v2f A;
    A.x = (half == 0) ? u : 1.0f;   // K=0 : u          | K=2 : 1 (dir z)
    A.y = (half == 0) ? v : 0.0f;   // K=1 : v          | K=3 : 0 (padding)

    // B: B[k][n] = R[n][k] (n<3), zero elsewhere
    const float* R = cam + b * 16;
    float b0 = 0.0f, b1 = 0.0f;
    if (nl < 3) {
        if (half == 0) { b0 = R[nl * 4 + 0]; b1 = R[nl * 4 + 1]; } // K=0,1
        else           { b0 = R[nl * 4 + 2]; b1 = 0.0f;         } // K=2,3
    }
    v2f Bm; Bm.x = b0; Bm.y = b1;

    v8f Cz = {0.f, 0.f, 0.f, 0.f, 0.f, 0.f, 0.f, 0.f};
    v8f Dm = __builtin_amdgcn_wmma_f32_16x16x4_f32(
        false, A, false, Bm, (short)0, Cz, false, false);

    // D extraction: lane (half,nl) VGPR r holds D[r + 8*half][nl]
    if (nl < 3) {
        float* outb = rdirs + (size_t)b * (HW * HW) * 3;
        int mbase = half * 8;
#pragma unroll
        for (int r = 0; r < 8; ++r) {
            outb[(size_t)(ray0 + mbase + r) * 3 + nl] = Dm[r];
        }
    }
}

// ---------------------------------------------------------------------------
// Stage 3: raymarch.  Block = 32 pixels (lanes) x 8 segments (waves).
// Each thread: 16 samples, segment-local exclusive-prefix accumulation
// (part[c] += rv[c+1]*cum; cum *= occ).  Exact combine across segments via
// LDS: out[c] = sum_g part_g[c] * prod_{g'<g} P_g'.
// ---------------------------------------------------------------------------
#define CORNER(DZ, DY, DX)                                                    \
    {                                                                         \
        float wk = (DZ ? fz : 1.0f - fz) * (DY ? fy : 1.0f - fy) *            \
                   (DX ? fx : 1.0f - fx);                                     \
        const float4* q = basep + ((DZ) * (PD * PD) + (DY) * PD + (DX)) * 4;  \
        float4 q0 = q[0], q1 = q[1], q2 = q[2], q3 = q[3];                    \
        rv0  += wk * q0.x; rv1  += wk * q0.y; rv2  += wk * q0.z;              \
        rv3  += wk * q0.w; rv4  += wk * q1.x; rv5  += wk * q1.y;              \
        rv6  += wk * q1.z; rv7  += wk * q1.w; rv8  += wk * q2.x;              \
        rv9  += wk * q2.y; rv10 += wk * q2.z; rv11 += wk * q2.w;              \
        rv12 += wk * q3.x;                                                    \
    }

template <bool PACKED>
__global__ void raymarch_kernel(const float* __restrict__ vox,
                                const float* __restrict__ cam,
                                const float* __restrict__ packed,
                                const float* __restrict__ rdirs,
                                float* __restrict__ out) {
    const int lane = threadIdx.x & 31;
    const int seg  = threadIdx.x >> 5;
    const int pix  = blockIdx.x * 32 + lane;     // 0..NPIX-1
    const int b    = pix >> 14;                  // 16384 pixels per image
    const int hw   = pix & 16383;
    const int h    = hw >> 7, w = hw & 127;

    const float* cp = cam + b * 16;
    const float trx = cp[3], tryy = cp[7], trz = cp[11];

    float rx, ry, rz;
    if (PACKED) {
        const float* rd = rdirs + (size_t)pix * 3;
        rx = rd[0]; ry = rd[1]; rz = rd[2];
    } else {
        float u = ((float)w + 0.5f - 64.0f) * INV_FOCAL;
        float v = ((float)h + 0.5f - 64.0f) * INV_FOCAL;
        rx = cp[0] * u + cp[1] * v + cp[2];
        ry = cp[4] * u + cp[5] * v + cp[6];
        rz = cp[8] * u + cp[9] * v + cp[10];
    }

    float cum = 1.0f;
    float part[12];
#pragma unroll
    for (int c = 0; c < 12; ++c) part[c] = 0.0f;

#pragma unroll 2
    for (int j = 0; j < SEG_LEN; ++j) {
        const int   s = seg * SEG_LEN + j;
        const float t = T_NEAR + (float)s * T_STEP;
        const float wxp = t * rx + trx;
        const float wyp = t * ry + tryy;
        const float wzp = t * rz + trz;
        // coords(+1 for pad) = world*64 + 32.5, clipped to [0, 64.9999]
        const float cz = fminf(fmaxf(wzp * 64.0f + 32.5f, 0.0f), 64.9999f);
        const float cy = fminf(fmaxf(wyp * 64.0f + 32.5f, 0.0f), 64.9999f);
        const float cx = fminf(fmaxf(wxp * 64.0f + 32.5f, 0.0f), 64.9999f);
        const int iz = (int)cz, iy = (int)cy, ix = (int)cx;
        const float fz = cz - (float)iz;
        const float fy = cy - (float)iy;
        const float fx = cx - (float)ix;

        float rv0 = 0, rv1 = 0, rv2 = 0, rv3 = 0, rv4 = 0, rv5 = 0, rv6 = 0,
              rv7 = 0, rv8 = 0, rv9 = 0, rv10 = 0, rv11 = 0, rv12 = 0;

        if (PACKED) {
            const size_t cell = ((size_t)((b * PD + iz) * PD + iy) * PD + ix);
            const float4* basep = (const float4*)packed + cell * 4;
            CORNER(0, 0, 0) CORNER(0, 0, 1) CORNER(0, 1, 0) CORNER(0, 1, 1)
            CORNER(1, 0, 0) CORNER(1, 0, 1) CORNER(1, 1, 0) CORNER(1, 1, 1)

            // gfx1250 global_prefetch of next sample's cell (hide L2 latency)
            const float wz2 = wzp + T_STEP * rz;
            const float wy2 = wyp + T_STEP * ry;
            const float wx2 = wxp + T_STEP * rx;
            const int iz2 = (int)fminf(fmaxf(wz2 * 64.0f + 32.5f, 0.0f), 64.9999f);
            const int iy2 = (int)fminf(fmaxf(wy2 * 64.0f + 32.5f, 0.0f), 64.9999f);
            const int ix2 = (int)fminf(fmaxf(wx2 * 64.0f + 32.5f, 0.0f), 64.9999f);
            const char* pf = (const char*)packed +
                ((size_t)((b * PD + iz2) * PD + iy2) * PD + ix2) * 64;
            __builtin_prefetch(pf, 0, 0);
        } else {
            float rv[NCH];
#pragma unroll
            for (int c = 0; c < NCH; ++c) rv[c] = 0.0f;
#pragma unroll
            for (int k = 0; k < 8; ++k) {
                const int dz = (k >> 2) & 1, dy = (k >> 1) & 1, dx = k & 1;
                const int z = iz + dz, y = iy + dy, x = ix + dx; // padded coords
                const float wk = (dz ? fz : 1.0f - fz) * (dy ? fy : 1.0f - fy) *
                                 (dx ? fx : 1.0f - fx);
                if (z >= 1 && z <= VD && y >= 1 && y <= VD && x >= 1 && x <= VD) {
                    const float* src = vox +
                        ((((size_t)b * VD + (z - 1)) * VD + (y - 1)) * VD +
                         (x - 1)) * NCH;
#pragma unroll
                    for (int c = 0; c < NCH; ++c)
                        rv[c] += wk / (1.0f + expf(-src[c]));
                }
            }
            rv0 = rv[0];  rv1 = rv[1];  rv2 = rv[2];  rv3 = rv[3];
            rv4 = rv[4];  rv5 = rv[5];  rv6 = rv[6];  rv7 = rv[7];
            rv8 = rv[8];  rv9 = rv[9];  rv10 = rv[10]; rv11 = rv[11];
            rv12 = rv[12];
        }

        const float occ = rv0 + 1e-12f;
        part[0]  = fmaf(rv1,  cum, part[0]);
        part[1]  = fmaf(rv2,  cum, part[1]);
        part[2]  = fmaf(rv3,  cum, part[2]);
        part[3]  = fmaf(rv4,  cum, part[3]);
        part[4]  = fmaf(rv5,  cum, part[4]);
        part[5]  = fmaf(rv6,  cum, part[5]);
        part[6]  = fmaf(rv7,  cum, part[6]);
        part[7]  = fmaf(rv8,  cum, part[7]);
        part[8]  = fmaf(rv9,  cum, part[8]);
        part[9]  = fmaf(rv10, cum, part[9]);
        part[10] = fmaf(rv11, cum, part[10]);
        part[11] = fmaf(rv12, cum, part[11]);
        cum *= occ;
    }

    // Exact cross-segment combine in LDS.
    __shared__ float sP[SEGS][32];
    __shared__ float sPart[SEGS][32][12];
    sP[seg][lane] = cum;
#pragma unroll
    for (int c = 0; c < 12; ++c) sPart[seg][lane][c] = part[c];
    __syncthreads();

    if (seg == 0) {
        float cumP = 1.0f;
        float acc[12];
#pragma unroll
        for (int c = 0; c < 12; ++c) acc[c] = 0.0f;
#pragma unroll
        for (int g = 0; g < SEGS; ++g) {
#pragma unroll
            for (int c = 0; c < 12; ++c)
                acc[c] = fmaf(sPart[g][lane][c], cumP, acc[c]);
            cumP *= sP[g][lane];
        }
        float sum = 0.0f;
#pragma unroll
        for (int c = 0; c < 12; ++c) sum += acc[c];
        float* o = out + (size_t)pix * NCH;
        o[0] = 1.0f - sum;
#pragma unroll
        for (int c = 0; c < 12; ++c) o[1 + c] = acc[c];
    }
}

// ---------------------------------------------------------------------------
extern "C" void kernel_launch(void* const* d_in, const int* in_sizes, int n_in,
                              void* d_out, int out_size, void* d_ws,
                              size_t ws_size, hipStream_t stream) {
    (void)in_sizes; (void)n_in; (void)out_size;
    const float* vox = (const float*)d_in[0];   // (2,64,64,64,13) f32
    const float* cam = (const float*)d_in[1];   // (2,4,4) f32
    float* out = (float*)d_out;                 // (2,128,128,13) f32

    const size_t packedFloats = (size_t)BATCH * PD * PD * PD * PCH; // 9,199,872
    const size_t rdirFloats   = (size_t)NPIX * 3;                   //    98,304
    const size_t needed = (packedFloats + rdirFloats) * sizeof(float);

    if (ws_size >= needed) {
        float* packed = (float*)d_ws;
        float* rdirs  = packed + packedFloats;
        const int totalVox = BATCH * PD * PD * PD;                  // 574,992
        pack_kernel<<<(totalVox + 255) / 256, 256, 0, stream>>>(vox, packed,
                                                                totalVox);
        rdir_wmma_kernel<<<256, 256, 0, stream>>>(cam, rdirs);      // 2048 waves
        raymarch_kernel<true><<<NPIX / 32, 256, 0, stream>>>(vox, cam, packed,
                                                             rdirs, out);
    } else {
        raymarch_kernel<false><<<NPIX / 32, 256, 0, stream>>>(vox, cam, nullptr,
                                                              nullptr, out);
    }
}